// Net_74036646248705
// MI455X (gfx1250) — compile-verified
//
#include <hip/hip_runtime.h>

typedef float v2f __attribute__((ext_vector_type(2)));
typedef float v4f __attribute__((ext_vector_type(4)));
typedef float v8f __attribute__((ext_vector_type(8)));

#define BT     256   // threads per block (8 wave32)
#define NB     512   // blocks in stage 1 (partials in workspace)
#define NWAVES 8

// Ordered product of the BT per-thread 2x2 partials of this block, written to
// dst[0..3] (row-major) by tid 0. Tree levels use V_WMMA_F32_16X16X4_F32:
// one WMMA computes 8 independent 2x2 products packed block-diagonally.
__device__ __forceinline__ void tree_combine(float* __restrict__ dst,
                                             float m00, float m01,
                                             float m10, float m11)
{
    __shared__ float buf[2][BT * 4];
    const int tid  = threadIdx.x;
    const int wid  = tid >> 5;
    const int lane = tid & 31;

    buf[0][tid * 4 + 0] = m00; buf[0][tid * 4 + 1] = m01;
    buf[0][tid * 4 + 2] = m10; buf[0][tid * 4 + 3] = m11;
    __syncthreads();

    // Per-lane decomposition of the WMMA register layout (wave32):
    //  A 16x4 f32 : lanes 0-15 hold row M=lane, VGPR0=K0,VGPR1=K1; lanes 16-31 K2,K3
    //  B 4x16 f32 : lanes 0-15 hold col N=lane, VGPR0=K0,VGPR1=K1; lanes 16-31 K2,K3
    //  D 16x16 f32: VGPR r -> (M=r, N=lane) for lanes 0-15, (M=r+8, N=lane-16) hi lanes
    const int hi = lane >> 4;   // which K/M half this lane serves
    const int qn = lane & 15;   // row index (A), col index (B/D)
    const int il = qn >> 1;     // local pair id 0..7 within one WMMA
    const int rc = qn & 1;      // row within L (for A) / col within R (for B/D)
    const int g  = il & 1;      // K-group used by pair il
    const int sel   = il & 3;   // D-register pair selector for owner lanes
    const bool ownr = ((qn >> 3) == hi);

    int P = BT, src = 0;
    while (P > 1) {
        const float* q = buf[src];
        float*       o = buf[src ^ 1];
        const int numPairs = P >> 1;
        const int numInst  = (numPairs + 7) >> 3;

        for (int inst = wid; inst < numInst; inst += NWAVES) {
            const int pairIdx = (inst << 3) + il;
            const bool feed   = (pairIdx < numPairs) && (g == hi);
            const int  ML     = pairIdx << 1;          // left matrix index
            // Unconditional LDS loads (always in-bounds), then value-mask:
            // compiles to ds_load_b32 + v_cndmask, no exec save/restore.
            const float la0 = q[ML * 4 + rc * 2 + 0];  // L[rc][0]
            const float la1 = q[ML * 4 + rc * 2 + 1];  // L[rc][1]
            const float rb0 = q[ML * 4 + 4 + rc];      // R[0][rc]
            const float rb1 = q[ML * 4 + 6 + rc];      // R[1][rc]
            v2f a, b;
            a.x = feed ? la0 : 0.0f;
            a.y = feed ? la1 : 0.0f;
            b.x = feed ? rb0 : 0.0f;
            b.y = feed ? rb1 : 0.0f;
            v8f c = {0.f, 0.f, 0.f, 0.f, 0.f, 0.f, 0.f, 0.f};
            v8f d = __builtin_amdgcn_wmma_f32_16x16x4_f32(
                        false, a, false, b, (short)0, c, false, false);
            // Diagonal block of pair il: column qn needs D rows {2il, 2il+1},
            // which owner lanes hold in regs d[2*sel], d[2*sel+1].
            const float lo0 = (sel == 0) ? d[0] : d[2];
            const float lo1 = (sel == 0) ? d[1] : d[3];
            const float hi0 = (sel == 2) ? d[4] : d[6];
            const float hi1 = (sel == 2) ? d[5] : d[7];
            const float e0  = (sel < 2) ? lo0 : hi0;   // P[0][rc]
            const float e1  = (sel < 2) ? lo1 : hi1;   // P[1][rc]
            if (ownr && (pairIdx < numPairs)) {
                o[pairIdx * 4 + rc]     = e0;
                o[pairIdx * 4 + 2 + rc] = e1;
            }
        }
        __syncthreads();
        P = numPairs;
        src ^= 1;
    }

    if (tid == 0) {
        const float* q = buf[src];
        dst[0] = q[0]; dst[1] = q[1]; dst[2] = q[2]; dst[3] = q[3];
    }
}

// Stage 1: each thread folds a contiguous chunk of C matrices (streaming,
// non-temporal B128 loads), block tree-combines, writes block partial to ws.
__global__ __launch_bounds__(BT) void chain_stage1(const float* __restrict__ A,
                                                   float* __restrict__ ws,
                                                   int N, int C, float inv_n)
{
    const int t = blockIdx.x * BT + threadIdx.x;
    int i   = t * C;
    int end = i + C;
    if (end > N) end = N;

    float x00 = 1.f, x01 = 0.f, x10 = 0.f, x11 = 1.f;
    const v4f* Ap = (const v4f*)A;

#pragma unroll 8
    for (; i < end; ++i) {
        v4f a = __builtin_nontemporal_load(Ap + i);   // global_load_b128, NT
        const float b00 = fmaf(a.x, inv_n, 1.0f);
        const float b01 = a.y * inv_n;
        const float b10 = a.z * inv_n;
        const float b11 = fmaf(a.w, inv_n, 1.0f);
        const float y00 = fmaf(x00, b00, x01 * b10);
        const float y01 = fmaf(x00, b01, x01 * b11);
        const float y10 = fmaf(x10, b00, x11 * b10);
        const float y11 = fmaf(x10, b01, x11 * b11);
        x00 = y00; x01 = y01; x10 = y10; x11 = y11;
    }
    tree_combine(ws + blockIdx.x * 4, x00, x01, x10, x11);
}

// Stage 2: one block folds the NB ordered block partials.
__global__ __launch_bounds__(BT) void chain_stage2(const float* __restrict__ ws,
                                                   float* __restrict__ out, int nb)
{
    const int tid = threadIdx.x;
    const int per = nb / BT;          // NB/BT = 2
    float x00 = 1.f, x01 = 0.f, x10 = 0.f, x11 = 1.f;
    for (int j = tid * per; j < tid * per + per; ++j) {
        const float b00 = ws[j * 4 + 0], b01 = ws[j * 4 + 1];
        const float b10 = ws[j * 4 + 2], b11 = ws[j * 4 + 3];
        const float y00 = fmaf(x00, b00, x01 * b10);
        const float y01 = fmaf(x00, b01, x01 * b11);
        const float y10 = fmaf(x10, b00, x11 * b10);
        const float y11 = fmaf(x10, b01, x11 * b11);
        x00 = y00; x01 = y01; x10 = y10; x11 = y11;
    }
    tree_combine(out, x00, x01, x10, x11);
}

extern "C" void kernel_launch(void* const* d_in, const int* in_sizes, int n_in,
                              void* d_out, int out_size, void* d_ws, size_t ws_size,
                              hipStream_t stream) {
    const float* A   = (const float*)d_in[0];
    float*       out = (float*)d_out;
    float*       ws  = (float*)d_ws;   // NB * 4 floats = 8 KB

    const int N     = in_sizes[0] / 4;            // number of 2x2 matrices
    const int total = NB * BT;
    const int C     = (N + total - 1) / total;    // chunk per thread
    const float inv_n = 1.0f / (float)N;

    chain_stage1<<<dim3(NB), dim3(BT), 0, stream>>>(A, ws, N, C, inv_n);
    chain_stage2<<<dim3(1), dim3(BT), 0, stream>>>(ws, out, NB);
}